// QuantumLayer_23278722744726
// MI455X (gfx1250) — compile-verified
//
#include <hip/hip_runtime.h>

#define NQ 14
#define NL 6
#define NSTATE 16384   // 2^14 amplitudes (real)
#define TPB 256
#define EPT (NSTATE / TPB)  // 64 elements per thread

typedef float v2f __attribute__((ext_vector_type(2)));
typedef float v8f __attribute__((ext_vector_type(8)));

// Composite read-index map of the CNOT ring CNOT(0,1)..CNOT(12,13),CNOT(13,0).
// Wire w <-> bit (13-w). new[i] = old[f(i)], f = sigma_1 o ... o sigma_14
// (innermost = CNOT(13,0) applied first). Linear over GF(2).
__device__ __forceinline__ unsigned cnot_perm(unsigned y) {
  y ^= (y & 1u) << 13;                       // CNOT(13,0): bit0 -> flip bit13
#pragma unroll
  for (int k = 1; k <= 13; ++k)              // CNOT(13-k, 14-k): bitk -> bit(k-1)
    y ^= ((y >> k) & 1u) << (k - 1);
  return y;
}

// RY(theta): [[c,-s],[s,c]] with half angles
__device__ __forceinline__ float ry_elem(float c, float s, int rb, int cb) {
  return (rb == cb) ? c : ((rb == 0) ? -s : s);
}

// Build 16x16 U = R(t3) (x) R(t2) (x) R(t1) (x) R(t0) into LDS (row-major).
// t3 pairs with the MSB of the 4-bit group. Angle 0 == identity factor.
__device__ __forceinline__ void build_u(float* __restrict__ U,
                                        float t3, float t2, float t1, float t0,
                                        int tid) {
  float s3, c3, s2, c2, s1, c1, s0, c0;
  __sincosf(t3 * 0.5f, &s3, &c3);
  __sincosf(t2 * 0.5f, &s2, &c2);
  __sincosf(t1 * 0.5f, &s1, &c1);
  __sincosf(t0 * 0.5f, &s0, &c0);
  const int r = tid >> 4, c = tid & 15;
  float e = ry_elem(c3, s3, (r >> 3) & 1, (c >> 3) & 1);
  e *= ry_elem(c2, s2, (r >> 2) & 1, (c >> 2) & 1);
  e *= ry_elem(c1, s1, (r >> 1) & 1, (c >> 1) & 1);
  e *= ry_elem(c0, s0, r & 1, c & 1);
  U[tid] = e;
}

// Apply U (16x16) to the 4-bit group at bit position S, in place in LDS.
// State index: idx = (col>>S)<<(S+4) | k<<S | (col & ((1<<S)-1)), col in [0,1024).
// D = U x S(16x1024) as 64 column tiles of 16; K=16 done as 4 WMMA k-chunks.
// wv must be a scalar (wave-uniform) value so the tile loop is scalar-controlled
// and EXEC stays all-1s across every WMMA (ISA requirement).
template <int S>
__device__ __forceinline__ void apply_group_gate(float* __restrict__ st,
                                                 const float* __restrict__ U,
                                                 int lane, int wv) {
  const int khalf = lane >> 4;       // lanes 16-31 hold K+2 / rows M+8
  const int nl = lane & 15;
  // A fragment (16x4 f32): lane m=nl, VGPR0 = K(2*khalf), VGPR1 = K(2*khalf+1)
  v2f A[4];
#pragma unroll
  for (int kc = 0; kc < 4; ++kc) {
    A[kc].x = U[nl * 16 + kc * 4 + 2 * khalf + 0];
    A[kc].y = U[nl * 16 + kc * 4 + 2 * khalf + 1];
  }
  const unsigned lowmask = (1u << S) - 1u;
  // Unroll by 2: two independent accumulator chains per iteration for ILP.
#pragma unroll 2
  for (int tile = wv; tile < 64; tile += 8) {
    const unsigned c = (unsigned)tile * 16u + (unsigned)nl;
    const unsigned base = ((c >> S) << (S + 4)) | (c & lowmask);
    // B fragment (4x16 f32): lane n=nl, VGPR0 = K(2*khalf) row, VGPR1 = next
    v2f B[4];
#pragma unroll
    for (int kc = 0; kc < 4; ++kc) {
      const unsigned k0 = (unsigned)(kc * 4 + 2 * khalf);
      B[kc].x = st[base + (k0 << S)];
      B[kc].y = st[base + ((k0 + 1u) << S)];
    }
    v8f acc = {0.f, 0.f, 0.f, 0.f, 0.f, 0.f, 0.f, 0.f};
#pragma unroll
    for (int kc = 0; kc < 4; ++kc)
      acc = __builtin_amdgcn_wmma_f32_16x16x4_f32(false, A[kc], false, B[kc],
                                                  (short)0, acc, false, false);
    // D: VGPR r holds row (r + 8*khalf) of column c
#pragma unroll
    for (int r = 0; r < 8; ++r)
      st[base + ((unsigned)(r + 8 * khalf) << S)] = acc[r];
  }
}

// Apply RY on all 14 wires given 14 angles (expects barrier before call).
__device__ __forceinline__ void apply_ry_all(float* __restrict__ st,
                                             float* __restrict__ U,
                                             float a0, float a1, float a2, float a3,
                                             float a4, float a5, float a6, float a7,
                                             float a8, float a9, float a10, float a11,
                                             float a12, float a13,
                                             int tid, int lane, int wv) {
  build_u(U, a0, a1, a2, a3, tid);          // wires 0-3  -> bits 13-10
  __syncthreads();
  apply_group_gate<10>(st, U, lane, wv);
  __syncthreads();
  build_u(U, a4, a5, a6, a7, tid);          // wires 4-7  -> bits 9-6
  __syncthreads();
  apply_group_gate<6>(st, U, lane, wv);
  __syncthreads();
  build_u(U, a8, a9, a10, a11, tid);        // wires 8-11 -> bits 5-2
  __syncthreads();
  apply_group_gate<2>(st, U, lane, wv);
  __syncthreads();
  build_u(U, 0.f, 0.f, a12, a13, tid);      // I,I,w12,w13 -> bits 3-0
  __syncthreads();
  apply_group_gate<0>(st, U, lane, wv);
  __syncthreads();
}

__global__ __launch_bounds__(TPB) void qsim_kernel(const float* __restrict__ x,
                                                   const float* __restrict__ w,
                                                   float* __restrict__ out) {
  __shared__ float st[NSTATE];     // 64 KB state vector (real amplitudes)
  __shared__ float U[256];         // current 16x16 group operator
  __shared__ float zacc[NQ];
  __shared__ unsigned ftab[64];    // f(j<<8) for the linear CNOT-ring permutation

  const int tid = threadIdx.x;
  const int lane = tid & 31;
  // Wave index as an explicit scalar -> scalar loop control in the gate loops.
  const int wv = __builtin_amdgcn_readfirstlane(tid >> 5);
  const int b = blockIdx.x;

  // init |0...0>
#pragma unroll
  for (int j = 0; j < EPT; ++j) st[tid + TPB * j] = 0.0f;
  if (tid == 0) st[0] = 1.0f;
  if (tid < NQ) zacc[tid] = 0.0f;
  if (tid < 64) ftab[tid] = cnot_perm((unsigned)tid << 8);
  const unsigned fbase = cnot_perm((unsigned)tid);

  const float* xb = x + (size_t)b * NQ;
  __syncthreads();

  // input encoding: RY(x[b,i]) on wire i
  apply_ry_all(st, U,
               xb[0], xb[1], xb[2], xb[3], xb[4], xb[5], xb[6], xb[7],
               xb[8], xb[9], xb[10], xb[11], xb[12], xb[13],
               tid, lane, wv);

  for (int layer = 0; layer < NL; ++layer) {
    // CNOT ring as one permutation: st[i] <- st[f(i)], f linear over GF(2)
    float tmp[EPT];
#pragma unroll
    for (int j = 0; j < EPT; ++j) tmp[j] = st[fbase ^ ftab[j]];
    __syncthreads();
#pragma unroll
    for (int j = 0; j < EPT; ++j) st[tid + TPB * j] = tmp[j];
    __syncthreads();

    const float* wl = w + layer * NQ;
    apply_ry_all(st, U,
                 wl[0], wl[1], wl[2], wl[3], wl[4], wl[5], wl[6], wl[7],
                 wl[8], wl[9], wl[10], wl[11], wl[12], wl[13],
                 tid, lane, wv);
  }

  // <Z_w> = sum_i (+-) state[i]^2, sign from bit (13-w)
  float zp[NQ];
#pragma unroll
  for (int q = 0; q < NQ; ++q) zp[q] = 0.0f;
#pragma unroll 4
  for (int j = 0; j < EPT; ++j) {
    const unsigned i = (unsigned)tid + TPB * j;
    const float v = st[i];
    const float p = v * v;
#pragma unroll
    for (int q = 0; q < NQ; ++q)
      zp[q] += ((i >> (13 - q)) & 1u) ? -p : p;
  }
#pragma unroll
  for (int q = 0; q < NQ; ++q) {
#pragma unroll
    for (int off = 16; off > 0; off >>= 1)
      zp[q] += __shfl_xor(zp[q], off, 32);
  }
  if (lane == 0) {
#pragma unroll
    for (int q = 0; q < NQ; ++q) atomicAdd(&zacc[q], zp[q]);
  }
  __syncthreads();
  if (tid < NQ) out[(size_t)b * NQ + tid] = zacc[tid];
}

extern "C" void kernel_launch(void* const* d_in, const int* in_sizes, int n_in,
                              void* d_out, int out_size, void* d_ws, size_t ws_size,
                              hipStream_t stream) {
  const float* x = (const float*)d_in[0];      // (batch, 14) float32
  const float* w = (const float*)d_in[1];      // (6, 14) float32
  float* out = (float*)d_out;                  // (batch, 14) float32
  const int batch = in_sizes[0] / NQ;
  qsim_kernel<<<batch, TPB, 0, stream>>>(x, w, out);
}